// CheckpointAttention_29764123361314
// MI455X (gfx1250) — compile-verified
//
#include <hip/hip_runtime.h>
#include <hip/hip_bf16.h>
#include <math.h>

// ---- problem constants (match reference setup_inputs) ----
#define BB   4096      // batch rows
#define QQ   1024      // pooled_h feature dim
#define KK   256       // key dim
#define NN   16384     // number of snapshots
#define FF   512       // embedding dim
#define TOPK 2

typedef __attribute__((ext_vector_type(2))) float v2f;
typedef __attribute__((ext_vector_type(8))) float v8f;

// =====================================================================
// Kernel 1: q' = (pooled_h @ Wq + bq) * (1/16)
// Folding the 1/sqrt(K) score scale into q so scores = q' @ keys^T.
// Block = 256 threads = 8 waves; each wave computes one 16x16 output
// tile via chained V_WMMA_F32_16X16X4_F32 over K=1024 (256 steps).
// Grid: (4096/16 row groups) x (256/128 col halves) = 512 blocks.
// =====================================================================
__global__ void __launch_bounds__(256)
qproj_wmma_kernel(const float* __restrict__ ph,   // [B, Q] row-major
                  const float* __restrict__ Wq,   // [Q, K] row-major
                  const float* __restrict__ bq,   // [K]
                  float* __restrict__ qout)       // [B, K] row-major
{
    const int blk   = blockIdx.x;
    const int mg    = blk >> 1;          // row-group 0..255
    const int ng    = blk & 1;           // col half 0..1
    const int mbase = mg * 16;

    const int tid  = threadIdx.x;
    const int wave = tid >> 5;           // 0..7
    const int lane = tid & 31;
    const int l16  = lane & 15;
    const int half = lane >> 4;

    const int nb = ng * 128 + wave * 16;          // col base of this wave's tile

    // A: pooled_h row for this lane, K-pair offset by half (ISA 16x4 f32 layout)
    const float* arow = ph + (size_t)(mbase + l16) * QQ + 2 * half;
    // B: Wq column for this lane (ISA 4x16 layout: n = lane%16, K = vgpr + 2*half)
    const float* bcol = Wq + nb + l16;

    v8f c = {};
#pragma unroll 4
    for (int k = 0; k < QQ; k += 4) {
        v2f a = *(const v2f*)(arow + k);                  // A[m][k0+2h], A[m][k0+2h+1]
        v2f b;
        b.x = bcol[(size_t)(k + 2 * half)     * KK];      // Wq[k0+2h    ][n]
        b.y = bcol[(size_t)(k + 2 * half + 1) * KK];      // Wq[k0+2h+1][n]
        c = __builtin_amdgcn_wmma_f32_16x16x4_f32(
                false, a, false, b, (short)0, c, false, false);
    }

    const float bias = bq[nb + l16];
#pragma unroll
    for (int v = 0; v < 8; ++v) {
        const int m = mbase + v + 8 * half;               // C layout: M = vgpr + 8*half
        qout[(size_t)m * KK + nb + l16] = (c[v] + bias) * 0.0625f;  // fold 1/sqrt(256)
    }
}

// =====================================================================
// Kernel 2: beta[row] = sigmoid(pooled_h[row] . Wb + bb)
// One 256-thread block per row.
// =====================================================================
__global__ void __launch_bounds__(256)
beta_kernel(const float* __restrict__ ph,   // [B, Q]
            const float* __restrict__ Wb,   // [Q, 1]
            const float* __restrict__ bb,   // [1]
            float* __restrict__ beta)       // [B]
{
    const int row = blockIdx.x;
    const int tid = threadIdx.x;
    float s = 0.0f;
    for (int i = tid; i < QQ; i += 256)
        s += ph[(size_t)row * QQ + i] * Wb[i];
    __shared__ float red[256];
    red[tid] = s;
    __syncthreads();
    for (int off = 128; off > 0; off >>= 1) {
        if (tid < off) red[tid] += red[tid + off];
        __syncthreads();
    }
    if (tid == 0) {
        const float x = red[0] + bb[0];
        beta[row] = 1.0f / (1.0f + __expf(-x));
    }
}

// =====================================================================
// Kernel 3: scores = q' @ keys^T (WMMA f32), running top-2 per row,
// 2-way softmax, scatter alpha, gather+mix 2 embedding rows * beta.
// Block = 256 threads = 8 waves, handles 16 rows. Wave w covers
// columns [w*2048, (w+1)*2048) in 128 tiles of 16. alpha region must
// be pre-zeroed (hipMemsetAsync before this kernel).
// =====================================================================
__global__ void __launch_bounds__(256)
scores_topk_mix_kernel(const float* __restrict__ q,     // [B, K] pre-scaled
                       const float* __restrict__ keys,  // [N, K]
                       const float* __restrict__ embs,  // [N, F]
                       const float* __restrict__ beta,  // [B]
                       float* __restrict__ out0,        // [B, F]
                       float* __restrict__ alpha)       // [B, N] pre-zeroed
{
    const int mbase = blockIdx.x * 16;
    const int tid   = threadIdx.x;
    const int wave  = tid >> 5;
    const int lane  = tid & 31;
    const int l16   = lane & 15;
    const int half  = lane >> 4;

    __shared__ float cval[16][256];
    __shared__ int   cidx[16][256];
    __shared__ float s_a0[16], s_a1[16], s_bt[16];
    __shared__ int   s_i0[16], s_i1[16];

    float best[8], second[8];
    int   bidx[8], sidx[8];
#pragma unroll
    for (int v = 0; v < 8; ++v) {
        best[v] = -INFINITY; second[v] = -INFINITY;
        bidx[v] = 0;         sidx[v]   = 0;
    }

    const float* arow = q + (size_t)(mbase + l16) * KK + 2 * half;
    const int nstart = wave * (NN / 8);                  // 2048 cols per wave

    for (int nt = 0; nt < (NN / 8) / 16; ++nt) {         // 128 tiles
        const int nb = nstart + nt * 16;
        const float* brow = keys + (size_t)(nb + l16) * KK + 2 * half;

        v8f c = {};
#pragma unroll 4
        for (int k = 0; k < KK; k += 4) {
            v2f a = *(const v2f*)(arow + k);             // q'[m][k-pair]
            v2f b = *(const v2f*)(brow + k);             // keys[n][k-pair]
            c = __builtin_amdgcn_wmma_f32_16x16x4_f32(
                    false, a, false, b, (short)0, c, false, false);
        }

        const int col = nb + l16;                        // C layout: N = lane%16
#pragma unroll
        for (int v = 0; v < 8; ++v) {
            const float val = c[v];
            if (val > best[v]) {
                second[v] = best[v]; sidx[v] = bidx[v];
                best[v]   = val;     bidx[v] = col;
            } else if (val > second[v]) {
                second[v] = val;     sidx[v] = col;
            }
        }
    }

    // Deposit per-lane top-2 candidates: row r gets 8 waves * 16 lanes * 2 = 256.
#pragma unroll
    for (int v = 0; v < 8; ++v) {
        const int r    = v + 8 * half;                   // C layout: M = vgpr + 8*half
        const int slot = wave * 32 + l16 * 2;
        cval[r][slot]     = best[v];   cidx[r][slot]     = bidx[v];
        cval[r][slot + 1] = second[v]; cidx[r][slot + 1] = sidx[v];
    }
    __syncthreads();

    // One thread per row: reduce 256 candidates -> global top-2, softmax, scatter.
    if (tid < 16) {
        float b0 = -INFINITY, b1 = -INFINITY;
        int   i0 = 0, i1 = 0;
        for (int s = 0; s < 256; ++s) {
            const float val = cval[tid][s];
            const int   ix  = cidx[tid][s];
            if (val > b0)      { b1 = b0; i1 = i0; b0 = val; i0 = ix; }
            else if (val > b1) { b1 = val; i1 = ix; }
        }
        const float e  = __expf(b1 - b0);                // <= 1, stable
        const float a0 = 1.0f / (1.0f + e);
        const float a1 = 1.0f - a0;
        const int row = mbase + tid;
        alpha[(size_t)row * NN + i0] = a0;
        alpha[(size_t)row * NN + i1] = a1;
        s_a0[tid] = a0; s_a1[tid] = a1;
        s_i0[tid] = i0; s_i1[tid] = i1;
        s_bt[tid] = beta[row];
    }
    __syncthreads();

    // Cooperative mix: out0[row] = (a0*embs[i0] + a1*embs[i1]) * beta[row]
    for (int e = tid; e < 16 * FF; e += 256) {
        const int r = e >> 9;                            // FF == 512
        const int f = e & (FF - 1);
        const float m = s_a0[r] * embs[(size_t)s_i0[r] * FF + f]
                      + s_a1[r] * embs[(size_t)s_i1[r] * FF + f];
        out0[(size_t)(mbase + r) * FF + f] = m * s_bt[r];
    }
}

// =====================================================================
// Host launcher
// =====================================================================
extern "C" void kernel_launch(void* const* d_in, const int* in_sizes, int n_in,
                              void* d_out, int out_size, void* d_ws, size_t ws_size,
                              hipStream_t stream) {
    const float* ph   = (const float*)d_in[0];   // pooled_h      [B,Q]
    const float* keys = (const float*)d_in[1];   // snapshot_keys [N,K]
    const float* embs = (const float*)d_in[2];   // snapshot_embs [N,F]
    const float* Wq   = (const float*)d_in[3];   // [Q,K]
    const float* bq   = (const float*)d_in[4];   // [K]
    const float* Wb   = (const float*)d_in[5];   // [Q,1]
    const float* bb   = (const float*)d_in[6];   // [1]
    // d_in[7] = fallback (unused by reference math), d_in[8] = topk (fixed 2)

    float* out0  = (float*)d_out;                       // [B,F]
    float* alpha = out0 + (size_t)BB * FF;              // [B,N]

    float* q_ws  = (float*)d_ws;                        // [B,K]  = 4 MB
    float* b_ws  = q_ws + (size_t)BB * KK;              // [B]    = 16 KB

    // 1) q' = (pooled_h @ Wq + bq) / 16   (WMMA f32)
    qproj_wmma_kernel<<<(BB / 16) * (KK / 128), 256, 0, stream>>>(ph, Wq, bq, q_ws);

    // 2) beta = sigmoid(pooled_h @ Wb + bb)
    beta_kernel<<<BB, 256, 0, stream>>>(ph, Wb, bb, b_ws);

    // 3) zero the (mostly-zero) alpha output; kernel boundary orders the
    //    later scatter of the 2 nonzeros/row after this memset.
    hipMemsetAsync(alpha, 0, (size_t)BB * NN * sizeof(float), stream);

    // 4) scores (WMMA f32) + top-2 + softmax + alpha scatter + embedding mix
    scores_topk_mix_kernel<<<BB / 16, 256, 0, stream>>>(q_ws, keys, embs, b_ws,
                                                        out0, alpha);
}